// NonLocalAttention_13804024889577
// MI455X (gfx1250) — compile-verified
//
#include <hip/hip_runtime.h>
#include <hip/hip_bf16.h>
#include <math.h>

// Problem constants (from reference)
#define T_    3
#define DIM_  256
#define H_    64
#define W_    64
#define HW_   4096
#define THW_  12288
#define NHD_  4
#define CH_   64
#define PS_   7
#define WS_   9
#define WSH_  4
#define KS_   16
#define KA_   8
#define Q_    3072
#define SCALE_ 0.125f

typedef float v2f __attribute__((ext_vector_type(2)));
typedef float v8f __attribute__((ext_vector_type(8)));

__device__ __forceinline__ int reflect_i(int i, int n) {
  i = (i < 0) ? -i : i;
  return (i >= n) ? (2 * (n - 1) - i) : i;
}

// ---------------------------------------------------------------------------
// 1) Depthwise 3x3 (zero-pad SAME) for q/k/v in one pass over vid
// ---------------------------------------------------------------------------
__global__ void k_dwconv(const float* __restrict__ vid,
                         const float* __restrict__ qdw, const float* __restrict__ kdw,
                         const float* __restrict__ vdw,
                         float* __restrict__ tq, float* __restrict__ tk, float* __restrict__ tv) {
  int idx = blockIdx.x * blockDim.x + threadIdx.x;
  if (idx >= T_ * DIM_ * HW_) return;
  int x = idx & 63, y = (idx >> 6) & 63;
  int c = (idx >> 12) & 255;
  int t = idx >> 20;
  const float* src = vid + (size_t)t * (DIM_ * HW_) + (size_t)c * HW_;
  float aq = 0.f, ak = 0.f, av = 0.f;
#pragma unroll
  for (int ky = 0; ky < 3; ++ky) {
    int iy = y + ky - 1;
    if (iy < 0 || iy >= H_) continue;
#pragma unroll
    for (int kx = 0; kx < 3; ++kx) {
      int ix = x + kx - 1;
      if (ix < 0 || ix >= W_) continue;
      float val = src[iy * W_ + ix];
      int wi = c * 9 + ky * 3 + kx;
      aq = fmaf(val, qdw[wi], aq);
      ak = fmaf(val, kdw[wi], ak);
      av = fmaf(val, vdw[wi], av);
    }
  }
  tq[idx] = aq; tk[idx] = ak; tv[idx] = av;
}

// ---------------------------------------------------------------------------
// 2) Pointwise 1x1 conv as fp32 WMMA GEMM: Y[m,g] = sum_k W[m,k]*X[k,g]
//    M=K=256, N=T*HW; one wave per 16x16 tile; V_WMMA_F32_16X16X4_F32.
//    X/Y layout: [t][k][pix] ; 16-wide column tiles never straddle t (4096%16==0)
// ---------------------------------------------------------------------------
__global__ void k_gemm_pw(const float* __restrict__ Wm, const float* __restrict__ bias,
                          const float* __restrict__ X, float* __restrict__ Y, float scale) {
  const int lane = threadIdx.x;        // 0..31, EXEC all ones
  const int half = lane >> 4;
  const int l    = lane & 15;
  const int m_base = blockIdx.y << 4;
  const int g      = (blockIdx.x << 4) + l;
  const int t   = g >> 12;
  const int pix = g & 4095;
  const float* Xt = X + (size_t)t * (DIM_ * HW_) + pix;
  v8f acc = {};
  for (int k0 = 0; k0 < DIM_; k0 += 4) {
    const int kk = k0 + (half << 1);
    // A 16x4 fp32 layout: lanes 0-15 -> M=l, K=kk,kk+1 ; lanes 16-31 -> K=kk(=k0+2),kk+1
    v2f a = *(const v2f*)(Wm + (size_t)(m_base + l) * DIM_ + kk);
    v2f b;
    b.x = Xt[(size_t)kk * HW_];
    b.y = Xt[(size_t)(kk + 1) * HW_];
    acc = __builtin_amdgcn_wmma_f32_16x16x4_f32(false, a, false, b, (short)0, acc,
                                                false, false);
  }
  float* Yt = Y + (size_t)t * (DIM_ * HW_) + pix;
#pragma unroll
  for (int r = 0; r < 8; ++r) {
    const int m = m_base + r + (half << 3);   // C/D: VGPR r -> M=r (lanes<16) / r+8
    Yt[(size_t)m * HW_] = (acc[r] + bias[m]) * scale;
  }
}

// ---------------------------------------------------------------------------
// 3) Per-pixel q.k dots for all 81 offsets (reflect on k index)
//    dall[((off*4+h)*3+t)*4096 + pix]
// ---------------------------------------------------------------------------
__global__ void k_dall(const float* __restrict__ q, const float* __restrict__ k,
                       float* __restrict__ dall) {
  int idx = blockIdx.x * blockDim.x + threadIdx.x;
  if (idx >= 81 * NHD_ * T_ * HW_) return;
  int pix = idx & 4095;
  int t = (idx >> 12) % 3;
  int h = (idx / (3 * 4096)) & 3;
  int off = idx / (12 * 4096);
  int dy = off / WS_ - WSH_, dx = off % WS_ - WSH_;
  int y = pix >> 6, x = pix & 63;
  int ry = reflect_i(y + dy, H_), rx = reflect_i(x + dx, W_);
  const float* qp = q + (size_t)t * (DIM_ * HW_) + (size_t)(h * CH_) * HW_ + pix;
  const float* kp = k + (size_t)t * (DIM_ * HW_) + (size_t)(h * CH_) * HW_ + ry * W_ + rx;
  float s = 0.f;
#pragma unroll 4
  for (int c = 0; c < CH_; ++c) s = fmaf(qp[c * HW_], kp[c * HW_], s);
  dall[idx] = s;
}

// ---------------------------------------------------------------------------
// 4) 7x7 reflect box-sum evaluated at stride-2 grid queries
//    dists[(h*Q+q)*81 + off]
// ---------------------------------------------------------------------------
__global__ void k_boxsum(const float* __restrict__ dall, float* __restrict__ dists) {
  int idx = blockIdx.x * blockDim.x + threadIdx.x;
  if (idx >= NHD_ * Q_ * 81) return;
  int off = idx % 81;
  int q = (idx / 81) % Q_;
  int h = idx / (81 * Q_);
  int t = q >> 10, r = q & 1023;
  int gy = (r >> 5) << 1, gx = (r & 31) << 1;
  const float* dp = dall + (size_t)((off * NHD_ + h) * T_ + t) * HW_;
  float s = 0.f;
  for (int py = 0; py < PS_; ++py) {
    int yy = reflect_i(gy + py, H_) * W_;
    for (int px = 0; px < PS_; ++px)
      s += dp[yy + reflect_i(gx + px, W_)];
  }
  dists[idx] = s;
}

// ---------------------------------------------------------------------------
// 5) top-16 (ties -> lowest index, as jax top_k), softmax over 16, keep 8
// ---------------------------------------------------------------------------
__global__ void k_topk(const float* __restrict__ dists, float* __restrict__ attn,
                       int* __restrict__ inds) {
  int row = blockIdx.x * blockDim.x + threadIdx.x;
  if (row >= NHD_ * Q_) return;
  float v[81];
  const float* dp = dists + (size_t)row * 81;
  for (int o = 0; o < 81; ++o) v[o] = dp[o];
  float vals[KS_]; int id[KS_];
  for (int s = 0; s < KS_; ++s) {
    float m = -3.4e38f; int jm = 0;
    for (int o = 0; o < 81; ++o) { if (v[o] > m) { m = v[o]; jm = o; } }
    vals[s] = m; id[s] = jm; v[jm] = -3.4e38f;
  }
  float mx = vals[0], sum = 0.f, e[KS_];
  for (int s = 0; s < KS_; ++s) { e[s] = expf(vals[s] - mx); sum += e[s]; }
  float inv = 1.f / sum;
  for (int n = 0; n < KA_; ++n) {
    attn[row * KA_ + n] = e[n] * inv;
    inds[row * KA_ + n] = id[n];
  }
}

// ---------------------------------------------------------------------------
// 6) zero vacc+zv ; coverage counts
// ---------------------------------------------------------------------------
__global__ void k_zero(float* __restrict__ p, int n) {
  int i = blockIdx.x * blockDim.x + threadIdx.x;
  if (i < n) p[i] = 0.f;
}

__global__ void k_counts(float* __restrict__ zv) {
  int col = blockIdx.x * blockDim.x + threadIdx.x;
  if (col >= Q_ * PS_ * PS_) return;
  int q = col / 49, pp = col % 49, py = pp / 7, px = pp % 7;
  int t = q >> 10, r = q & 1023, yq = (r >> 5) << 1, xq = (r & 31) << 1;
  int lin = t * HW_ + reflect_i(yq + py, H_) * W_ + reflect_i(xq + px, W_);
  atomicAdd(&zv[lin], 1.0f);
}

// ---------------------------------------------------------------------------
// 7) attention-weighted gather of v patches -> pf[col(q,py,px)][dim(h*64+c)]
// ---------------------------------------------------------------------------
__global__ void k_pf(const float* __restrict__ v, const float* __restrict__ attn,
                     const int* __restrict__ inds, float* __restrict__ pf) {
  int col = blockIdx.x;      // 0..150527
  int dim = threadIdx.x;     // 0..255
  int h = dim >> 6;
  int q = col / 49, pp = col % 49, py = pp / 7, px = pp % 7;
  int t = q >> 10, r = q & 1023, yq = (r >> 5) << 1, xq = (r & 31) << 1;
  const float* vp = v + (size_t)t * (DIM_ * HW_) + (size_t)dim * HW_;
  int arow = (h * Q_ + q) * KA_;
  float acc = 0.f;
#pragma unroll
  for (int n = 0; n < KA_; ++n) {
    float a = attn[arow + n];
    int id = inds[arow + n];
    int yy = reflect_i(yq + id / WS_ - WSH_ + py, H_);
    int xx = reflect_i(xq + id % WS_ - WSH_ + px, W_);
    acc = fmaf(a, vp[yy * W_ + xx], acc);
  }
  pf[(size_t)col * DIM_ + dim] = acc;
}

// ---------------------------------------------------------------------------
// 8) Projection GEMM (fp32 WMMA) fused with scatter-add into vacc[c][t*HW+lin]
//    out[col, m] = sum_k pf[col,k]*proj_w[m,k] + b[m]; each lane owns one col.
// ---------------------------------------------------------------------------
__global__ void k_gemm_proj(const float* __restrict__ Wm, const float* __restrict__ bias,
                            const float* __restrict__ PF, float* __restrict__ vacc) {
  const int lane = threadIdx.x;
  const int half = lane >> 4;
  const int l    = lane & 15;
  const int m_base = blockIdx.y << 4;
  const int g      = (blockIdx.x << 4) + l;   // patch-element column
  const float* pfc = PF + (size_t)g * DIM_;
  v8f acc = {};
  for (int k0 = 0; k0 < DIM_; k0 += 4) {
    const int kk = k0 + (half << 1);
    v2f a = *(const v2f*)(Wm + (size_t)(m_base + l) * DIM_ + kk);
    v2f b = *(const v2f*)(pfc + kk);
    acc = __builtin_amdgcn_wmma_f32_16x16x4_f32(false, a, false, b, (short)0, acc,
                                                false, false);
  }
  int q = g / 49, pp = g % 49, py = pp / 7, px = pp % 7;
  int t = q >> 10, r2 = q & 1023, yq = (r2 >> 5) << 1, xq = (r2 & 31) << 1;
  int lin = t * HW_ + reflect_i(yq + py, H_) * W_ + reflect_i(xq + px, W_);
#pragma unroll
  for (int r = 0; r < 8; ++r) {
    int m = m_base + r + (half << 3);
    atomicAdd(&vacc[(size_t)m * THW_ + lin], acc[r] + bias[m]);
  }
}

// ---------------------------------------------------------------------------
// 9) normalize + layout to (B,T,DIM,H,W)
// ---------------------------------------------------------------------------
__global__ void k_final(const float* __restrict__ vacc, const float* __restrict__ zv,
                        float* __restrict__ out) {
  int idx = blockIdx.x * blockDim.x + threadIdx.x;
  if (idx >= T_ * DIM_ * HW_) return;
  int pix = idx & 4095;
  int c = (idx >> 12) & 255;
  int t = idx >> 20;
  out[idx] = vacc[(size_t)c * THW_ + t * HW_ + pix] / zv[t * HW_ + pix];
}

// ---------------------------------------------------------------------------
extern "C" void kernel_launch(void* const* d_in, const int* in_sizes, int n_in,
                              void* d_out, int out_size, void* d_ws, size_t ws_size,
                              hipStream_t stream) {
  const float* vid    = (const float*)d_in[0];
  const float* q_dw   = (const float*)d_in[1];
  const float* q_pw   = (const float*)d_in[2];
  const float* q_b    = (const float*)d_in[3];
  const float* k_dw   = (const float*)d_in[4];
  const float* k_pw   = (const float*)d_in[5];
  const float* k_b    = (const float*)d_in[6];
  const float* v_dw   = (const float*)d_in[7];
  const float* v_pw   = (const float*)d_in[8];
  const float* v_b    = (const float*)d_in[9];
  const float* proj_w = (const float*)d_in[10];
  const float* proj_b = (const float*)d_in[11];
  float* ws = (float*)d_ws;

  // workspace layout (float offsets), total 65,740,800 floats (~263 MB)
  float* tq    = ws + 0;          // 3,145,728
  float* tk    = ws + 3145728;
  float* tv    = ws + 6291456;
  float* qb    = ws + 9437184;
  float* kb    = ws + 12582912;
  float* vb    = ws + 15728640;
  float* dall  = ws + 18874368;   // 3,981,312
  float* dists = ws + 22855680;   //   995,328
  float* attn  = ws + 23851008;   //    98,304
  int*   inds  = (int*)(ws + 23949312); // 98,304
  float* pf    = ws + 24047616;   // 38,535,168
  float* vacc  = ws + 62582784;   // 3,145,728
  float* zv    = ws + 65728512;   //    12,288 (contiguous after vacc)

  // 1) depthwise convs
  k_dwconv<<<(T_*DIM_*HW_ + 255) / 256, 256, 0, stream>>>(vid, q_dw, k_dw, v_dw, tq, tk, tv);

  // 2) pointwise 1x1 convs as fp32 WMMA GEMMs (scale folded into q)
  dim3 gq(768, 16);
  k_gemm_pw<<<gq, 32, 0, stream>>>(q_pw, q_b, tq, qb, SCALE_);
  k_gemm_pw<<<gq, 32, 0, stream>>>(k_pw, k_b, tk, kb, 1.0f);
  k_gemm_pw<<<gq, 32, 0, stream>>>(v_pw, v_b, tv, vb, 1.0f);

  // 3) per-offset dots, 4) box-sums at query grid
  k_dall<<<(81*NHD_*T_*HW_) / 256, 256, 0, stream>>>(qb, kb, dall);
  k_boxsum<<<(NHD_*Q_*81) / 128, 128, 0, stream>>>(dall, dists);

  // 5) top-k + softmax
  k_topk<<<(NHD_*Q_ + 63) / 64, 64, 0, stream>>>(dists, attn, inds);

  // 6) zero accumulators (vacc + zv contiguous), coverage counts
  k_zero<<<(3145728 + 12288 + 255) / 256, 256, 0, stream>>>(vacc, 3145728 + 12288);
  k_counts<<<(Q_*49 + 255) / 256, 256, 0, stream>>>(zv);

  // 7) weighted v-patch gather -> pf
  k_pf<<<Q_*49, 256, 0, stream>>>(vb, attn, inds, pf);

  // 8) projection GEMM (WMMA) + fused atomic scatter
  dim3 gp(9408, 16);
  k_gemm_proj<<<gp, 32, 0, stream>>>(proj_w, proj_b, pf, vacc);

  // 9) normalize
  k_final<<<(T_*DIM_*HW_ + 255) / 256, 256, 0, stream>>>(vacc, zv, (float*)d_out);
}